// HistogramOfFeaturesModel_41566693491164
// MI455X (gfx1250) — compile-verified
//
#include <hip/hip_runtime.h>
#include <hip/hip_bf16.h>

// Problem dims (from reference)
#define B_ROWS 4096
#define FEAT   16384
#define OUT_N  64
#define BINS   128
#define KDIM   257          // 2*BINS+1
#define KPAD   260          // padded to multiple of 4 for 16x16x4 WMMA

typedef __attribute__((ext_vector_type(2))) float v2f;
typedef __attribute__((ext_vector_type(8))) float v8f;

// ---------------------------------------------------------------------------
// Kernel 0: build Wpad[k][n] (K-major, zero-padded K=257..259) from W[n][k]
// ---------------------------------------------------------------------------
__global__ void wpad_kernel(const float* __restrict__ W, float* __restrict__ Wpad) {
    int i = blockIdx.x * blockDim.x + threadIdx.x;        // i = k*64 + n
    if (i >= KPAD * OUT_N) return;
    int k = i / OUT_N;
    int n = i - k * OUT_N;
    Wpad[i] = (k < KDIM) ? W[n * KDIM + k] : 0.0f;
}

// ---------------------------------------------------------------------------
// Kernel 1: per-row histogram features. One workgroup (512 thr = 16 waves)
// per row. One single HBM pass: 32 floats/thread kept in registers.
// ---------------------------------------------------------------------------
__global__ void hist_kernel(const float* __restrict__ x, float* __restrict__ feats) {
    __shared__ unsigned s_hist[BINS];
    __shared__ float s_rmn[16];
    __shared__ float s_rmx[16];
    __shared__ float s_mn, s_width;

    const int t    = threadIdx.x;          // 0..511
    const int lane = t & 31;
    const int wave = t >> 5;               // 0..15
    const size_t row = blockIdx.x;

    if (t < BINS) s_hist[t] = 0u;

    // --- load 32 floats (8 x float4, coalesced) into registers, track min/max
    const float4* xrow = (const float4*)(x + row * (size_t)FEAT);
    float4 v[8];
    float mn =  3.4e38f;
    float mx = -3.4e38f;
#pragma unroll
    for (int j = 0; j < 8; ++j) {
        float4 q = xrow[t + j * 512];
        v[j] = q;
        mn = fminf(mn, fminf(fminf(q.x, q.y), fminf(q.z, q.w)));
        mx = fmaxf(mx, fmaxf(fmaxf(q.x, q.y), fmaxf(q.z, q.w)));
    }

    // --- wave32 reduction
#pragma unroll
    for (int off = 16; off > 0; off >>= 1) {
        mn = fminf(mn, __shfl_xor(mn, off, 32));
        mx = fmaxf(mx, __shfl_xor(mx, off, 32));
    }
    if (lane == 0) { s_rmn[wave] = mn; s_rmx[wave] = mx; }
    __syncthreads();

    if (t == 0) {
        float a = s_rmn[0], b = s_rmx[0];
#pragma unroll
        for (int w = 1; w < 16; ++w) { a = fminf(a, s_rmn[w]); b = fmaxf(b, s_rmx[w]); }
        s_mn = a;
        s_width = b - a;
    }
    __syncthreads();

    const float rmn   = s_mn;
    const float width = s_width;
    const float scale = (float)BINS / ((width == 0.0f) ? 1.0f : width);

    // --- bin from registers into LDS histogram
#pragma unroll
    for (int j = 0; j < 8; ++j) {
        float4 q = v[j];
        int i0 = (int)floorf((q.x - rmn) * scale);
        int i1 = (int)floorf((q.y - rmn) * scale);
        int i2 = (int)floorf((q.z - rmn) * scale);
        int i3 = (int)floorf((q.w - rmn) * scale);
        i0 = min(max(i0, 0), BINS - 1);
        i1 = min(max(i1, 0), BINS - 1);
        i2 = min(max(i2, 0), BINS - 1);
        i3 = min(max(i3, 0), BINS - 1);
        atomicAdd(&s_hist[i0], 1u);
        atomicAdd(&s_hist[i1], 1u);
        atomicAdd(&s_hist[i2], 1u);
        atomicAdd(&s_hist[i3], 1u);
    }
    __syncthreads();

    // --- emit padded 260-wide feature row
    if (t < KPAD) {
        float out;
        if (t < BINS) {
            out = (float)s_hist[t] * (1.0f / (float)FEAT);     // density (sum == FEAT)
        } else if (t < KDIM) {
            float tt = (float)(t - BINS) * (1.0f / (float)BINS);
            out = rmn + tt * width;                             // boundaries
        } else {
            out = 0.0f;                                         // K padding
        }
        feats[row * (size_t)KPAD + t] = out;
    }
}

// ---------------------------------------------------------------------------
// Kernel 2: out[4096,64] = feats[4096,260] @ Wpad[260,64] + b via
// V_WMMA_F32_16X16X4_F32. 256 threads = 8 waves; each wave owns one 16x16
// output tile (2 M-tiles x 4 N-tiles per block). Wpad staged in dynamic LDS.
// ---------------------------------------------------------------------------
__global__ void gemm_kernel(const float* __restrict__ feats,
                            const float* __restrict__ Wpad,
                            const float* __restrict__ bias,
                            float* __restrict__ out) {
    extern __shared__ float ldsW[];                     // KPAD*OUT_N = 16640 floats

    for (int i = threadIdx.x; i < KPAD * OUT_N; i += blockDim.x)
        ldsW[i] = Wpad[i];
    __syncthreads();

    const int lane  = threadIdx.x & 31;
    const int wave  = threadIdx.x >> 5;                 // 0..7
    const int mtile = blockIdx.x * 2 + (wave >> 2);     // 0..255
    const int ntile = wave & 3;                         // 0..3

    const int mrow = lane & 15;                         // M for A-frag
    const int koff = (lane < 16) ? 0 : 2;               // K split across lane halves
    const int n    = ntile * 16 + (lane & 15);          // N for B/C/D frags

    const float* arow = feats + (size_t)(mtile * 16 + mrow) * KPAD;

    v8f c = {};
#pragma unroll 5
    for (int k = 0; k < KPAD; k += 4) {
        // A 16x4: VGPR0 = K {0 | 2}, VGPR1 = K {1 | 3} across lane halves
        v2f a = *(const v2f*)(arow + k + koff);
        // B 4x16: same K split, row striped across lanes
        v2f bv;
        bv.x = ldsW[(k + koff + 0) * OUT_N + n];
        bv.y = ldsW[(k + koff + 1) * OUT_N + n];
        c = __builtin_amdgcn_wmma_f32_16x16x4_f32(
            /*neg_a=*/false, a, /*neg_b=*/false, bv,
            /*c_mod=*/(short)0, c, /*reuse_a=*/false, /*reuse_b=*/false);
    }

    const float bn = bias[n];
    const int mbase = mtile * 16 + ((lane < 16) ? 0 : 8);
#pragma unroll
    for (int v = 0; v < 8; ++v) {
        out[(size_t)(mbase + v) * OUT_N + n] = c[v] + bn;
    }
}

// ---------------------------------------------------------------------------
extern "C" void kernel_launch(void* const* d_in, const int* in_sizes, int n_in,
                              void* d_out, int out_size, void* d_ws, size_t ws_size,
                              hipStream_t stream) {
    const float* x = (const float*)d_in[0];   // [4096, 16384]
    const float* W = (const float*)d_in[1];   // [64, 257]
    const float* b = (const float*)d_in[2];   // [64]
    float* out = (float*)d_out;               // [4096, 64]

    float* feats = (float*)d_ws;                              // [4096, 260]
    float* Wpad  = feats + (size_t)B_ROWS * KPAD;             // [260, 64]

    // Kernel 0: pad/transpose W
    wpad_kernel<<<(KPAD * OUT_N + 255) / 256, 256, 0, stream>>>(W, Wpad);

    // Kernel 1: one workgroup per row, single HBM pass
    hist_kernel<<<B_ROWS, 512, 0, stream>>>(x, feats);

    // Kernel 2: WMMA GEMM, 8 tiles per block (2 M-tiles x 4 N-tiles)
    const int gemm_blocks = (B_ROWS / 16) / 2;                // 128
    const size_t lds_bytes = (size_t)KPAD * OUT_N * sizeof(float);  // 66560
    gemm_kernel<<<gemm_blocks, 256, lds_bytes, stream>>>(feats, Wpad, b, out);
}